// PrimeExpLSTM_62637803045437
// MI455X (gfx1250) — compile-verified
//
#include <hip/hip_runtime.h>
#include <hip/hip_bf16.h>
#include <math.h>

typedef __attribute__((ext_vector_type(2))) float v2f;
typedef __attribute__((ext_vector_type(8))) float v8f;

#define PD 6
#define NTOT 108
#define T_LEN 2048
#define B_TOT 4096
#define CLAMP_V 10.0f
#define CH 4            // time-steps per chunk

// log of first 6 primes: 2,3,5,7,11,13
__device__ __constant__ float c_logp[6] = {
    0.69314718f, 1.09861229f, 1.60943791f, 1.94591015f, 2.39789527f, 2.56494936f };

__device__ __forceinline__ float tanh_f(float v) {
#if __has_builtin(__builtin_amdgcn_tanhf)
    return __builtin_amdgcn_tanhf(v);   // v_tanh_f32 on gfx1250
#else
    return tanhf(v);
#endif
}

// Map WMMA row m (0..15) -> (gate index, unit, valid).
// gidx: 0=i, 1=f, 2=g, 3=o  (matches W_in [ii,if,ig,io], W_hid [hi,hf,hg,ho], bias order)
// Rows: 0-2 = i units, 3-5 = f units, 8-10 = o units, 11-13 = g units, rest zero.
// Sigmoid rows (i,f,o) are pre-scaled by 0.5 so sigmoid(p) = 0.5*tanh(d)+0.5 directly.
__device__ __forceinline__ void rowmap(int m, int& g, int& un, bool& valid) {
    if (m < 3)       { g = 0; un = m;      valid = true; }
    else if (m < 6)  { g = 1; un = m - 3;  valid = true; }
    else if (m < 8)  { g = 0; un = 0;      valid = false; }
    else if (m < 11) { g = 3; un = m - 8;  valid = true; }
    else if (m < 14) { g = 2; un = m - 11; valid = true; }
    else             { g = 0; un = 0;      valid = false; }
}

// Kernel 1: reconstruct the 108 weights and the MDL penalty.
__global__ void k_weights(const float* __restrict__ z, const float* __restrict__ u,
                          float* __restrict__ w, float* __restrict__ pen_out) {
    __shared__ float red[128];
    const int tid = threadIdx.x;
    float pen = 0.f;
    for (int n = tid; n < NTOT; n += blockDim.x) {
        float lm = 0.f, p = 0.f;
        for (int k = 0; k < PD; ++k) {
            float zz = z[n * PD + k];
            lm = fmaf(zz, c_logp[k], lm);
            p  = fmaf(fabsf(zz), c_logp[k], p);
        }
        lm = fminf(fmaxf(lm, -CLAMP_V), CLAMP_V);
        w[n] = tanhf(u[n]) * __expf(lm);
        pen += p;
    }
    red[tid] = pen;
    __syncthreads();
    for (int s = blockDim.x / 2; s > 0; s >>= 1) {
        if (tid < s) red[tid] += red[tid + s];
        __syncthreads();
    }
    if (tid == 0) *pen_out = red[0];
}

// Kernel 2: one wave (32 lanes) advances 16 batch sequences through all T steps.
// Per step: one v_wmma_f32_16x16x4_f32 computes all 12 gate pre-activations
// (rows = gate-units, cols = 16 batches, K = h state), with the one-hot input
// projection + biases folded into the C operand via a 3x16 LDS table.
// Sigmoid pre-scaling is folded into A/C; both lane halves redundantly compute
// the c/h update from symmetric xor-16 shuffles so the next B operand needs no
// extra shuffle. Time is chunked by 4 with x / C-tile prefetch.
__global__ void __launch_bounds__(32)
k_lstm(const int* __restrict__ x, const float* __restrict__ w, float* __restrict__ out) {
    __shared__ __align__(16) float tab[48];   // tab[x][row], x in 0..2, row in 0..15

    const int  lane  = threadIdx.x;
    const bool low   = lane < 16;
    const int  m     = lane & 15;
    const int  bBase = blockIdx.x * 16;

    // Build C table: tab[x][row] = scale * (W_in[g][x][u] + b_i[g][u] + b_h[g][u])
    for (int idx = lane; idx < 48; idx += 32) {
        int xv = idx / 16, row = idx % 16;
        int g, un; bool valid;
        rowmap(row, g, un, valid);
        float v = 0.f;
        if (valid) {
            const float scale = (g == 2) ? 1.0f : 0.5f;   // tanh rows unscaled
            v = scale * (w[g * 9 + xv * 3 + un]           // W_in[g][x][u]
                       + w[72 + g * 3 + un]               // b_i gate
                       + w[84 + g * 3 + un]);             // b_h gate
        }
        tab[idx] = v;
    }

    // A operand (16x4 f32, 2 VGPRs): lanes 0-15 hold K=0,1; lanes 16-31 hold K=2,3.
    // A[m][k] = scale * W_hid[g][k][u] (K=3 row is zero padding).
    int g, un; bool valid;
    rowmap(m, g, un, valid);
    v2f a; a[0] = 0.f; a[1] = 0.f;
    if (valid) {
        const float scale = (g == 2) ? 1.0f : 0.5f;
        const int k0 = low ? 0 : 2;
        const int k1 = low ? 1 : 3;
        a[0] = scale * w[36 + g * 9 + k0 * 3 + un];
        if (k1 < 3) a[1] = scale * w[36 + g * 9 + k1 * 3 + un];
    }

    // Output projection weights (broadcast scalars).
    const float wo00 = w[96],  wo01 = w[97],  wo02 = w[98];
    const float wo10 = w[99],  wo11 = w[100], wo12 = w[101];
    const float wo20 = w[102], wo21 = w[103], wo22 = w[104];
    const float bo0  = w[105], bo1  = w[106], bo2  = w[107];

    // d[3..5] post-activation: lower half -> sigmoid (0.5,0.5), upper half -> tanh (1,0)
    const float postM = low ? 0.5f : 1.0f;
    const float postA = low ? 0.5f : 0.0f;
    const int   rowOff = low ? 0 : 8;

    __syncthreads();

    const int* xrow = x + (size_t)(bBase + m) * T_LEN;
    float*     orow = out + (size_t)(bBase + m) * T_LEN * 3;

    // B operand (4x16 f32, 2 VGPRs): V0 = {h0 | h2}, V1 = {h1 | 0} across lane halves.
    v2f bv; bv[0] = 0.f; bv[1] = 0.f;
    float c0 = 0.f, c1 = 0.f, c2 = 0.f;   // cell state (tracked by BOTH halves)

    int4 xx = *reinterpret_cast<const int4*>(xrow);   // first chunk of x

    for (int t0 = 0; t0 < T_LEN; t0 += CH) {
        // Prefetch next chunk of x (clamped so the last chunk re-reads itself).
        const int tn = (t0 + CH < T_LEN) ? (t0 + CH) : t0;
        const int4 xx_next = *reinterpret_cast<const int4*>(xrow + tn);

        const int xts[CH] = { xx.x, xx.y, xx.z, xx.w };

        // Prefetch all C tiles for this chunk from LDS.
        float4 cta[CH], ctb[CH];
#pragma unroll
        for (int j = 0; j < CH; ++j) {
            const float4* t4p = reinterpret_cast<const float4*>(&tab[xts[j] * 16 + rowOff]);
            cta[j] = t4p[0];
            ctb[j] = t4p[1];
        }

        float lbuf[3 * CH];

#pragma unroll
        for (int j = 0; j < CH; ++j) {
            v8f cin = { cta[j].x, cta[j].y, cta[j].z, cta[j].w,
                        ctb[j].x, ctb[j].y, ctb[j].z, ctb[j].w };

            // D = (scaled Wh) * h + (scaled Win-row + biases)
            v8f d = __builtin_amdgcn_wmma_f32_16x16x4_f32(
                false, a, false, bv, (short)0, cin, false, false);

            // lanes 0-15: d[0..2]=I_pre/2, d[3..5]=F_pre/2 ;
            // lanes 16-31: d[0..2]=O_pre/2, d[3..5]=G_pre
            const float s0 = fmaf(tanh_f(d[0]), 0.5f, 0.5f);    // sigmoid: I / O
            const float s1 = fmaf(tanh_f(d[1]), 0.5f, 0.5f);
            const float s2 = fmaf(tanh_f(d[2]), 0.5f, 0.5f);
            const float t3 = fmaf(tanh_f(d[3]), postM, postA);  // F (low) / G (up)
            const float t4 = fmaf(tanh_f(d[4]), postM, postA);
            const float t5 = fmaf(tanh_f(d[5]), postM, postA);

            // Symmetric half-swap: gives lower lanes {G,O}, upper lanes {F,I}.
            const float t3w = __shfl_xor(t3, 16);
            const float t4w = __shfl_xor(t4, 16);
            const float t5w = __shfl_xor(t5, 16);
            const float s0w = __shfl_xor(s0, 16);
            const float s1w = __shfl_xor(s1, 16);
            const float s2w = __shfl_xor(s2, 16);

            const float F0 = low ? t3 : t3w,  G0 = low ? t3w : t3;
            const float F1 = low ? t4 : t4w,  G1 = low ? t4w : t4;
            const float F2 = low ? t5 : t5w,  G2 = low ? t5w : t5;
            const float I0 = low ? s0 : s0w,  O0 = low ? s0w : s0;
            const float I1 = low ? s1 : s1w,  O1 = low ? s1w : s1;
            const float I2 = low ? s2 : s2w,  O2 = low ? s2w : s2;

            // Both halves carry identical, valid c/h for their batch.
            c0 = fmaf(F0, c0, I0 * G0);
            c1 = fmaf(F1, c1, I1 * G1);
            c2 = fmaf(F2, c2, I2 * G2);
            const float tc0 = tanh_f(c0), tc1 = tanh_f(c1), tc2 = tanh_f(c2);
            const float h0 = O0 * tc0, h1 = O1 * tc1, h2 = O2 * tc2;

            lbuf[j * 3 + 0] = fmaf(h0, wo00, fmaf(h1, wo10, fmaf(h2, wo20, bo0)));
            lbuf[j * 3 + 1] = fmaf(h0, wo01, fmaf(h1, wo11, fmaf(h2, wo21, bo1)));
            lbuf[j * 3 + 2] = fmaf(h0, wo02, fmaf(h1, wo12, fmaf(h2, wo22, bo2)));

            // Next B operand needs NO shuffle: upper half owns a valid h2.
            bv[0] = low ? h0 : h2;
            bv[1] = low ? h1 : 0.0f;
        }

        // 12 floats = 3 aligned b128 stores per chunk (lanes 0-15 only).
        if (low) {
            float4* po = reinterpret_cast<float4*>(orow + (size_t)t0 * 3);
            po[0] = make_float4(lbuf[0], lbuf[1], lbuf[2],  lbuf[3]);
            po[1] = make_float4(lbuf[4], lbuf[5], lbuf[6],  lbuf[7]);
            po[2] = make_float4(lbuf[8], lbuf[9], lbuf[10], lbuf[11]);
        }

        xx = xx_next;
    }
}

extern "C" void kernel_launch(void* const* d_in, const int* in_sizes, int n_in,
                              void* d_out, int out_size, void* d_ws, size_t ws_size,
                              hipStream_t stream) {
    const int*   x = (const int*)d_in[0];
    const float* z = (const float*)d_in[1];
    const float* u = (const float*)d_in[2];
    float* out = (float*)d_out;
    float* w   = (float*)d_ws;                            // 108 reconstructed weights
    float* pen = out + (size_t)B_TOT * T_LEN * 3;         // scalar MDL penalty slot

    k_weights<<<1, 128, 0, stream>>>(z, u, w, pen);
    k_lstm<<<B_TOT / 16, 32, 0, stream>>>(x, w, out);
}